// MeshUnpool_8323646619909
// MI455X (gfx1250) — compile-verified
//
#include <hip/hip_runtime.h>

// MeshUnpool for MI455X (gfx1250, wave32).
// Kernel 1: single 1024-thread workgroup builds the fine->coarse source map
//           using LDS scans (O(N log N)), staging keep_idx via CDNA5 async
//           global->LDS loads (ASYNCcnt).
// Kernel 2: memory-bound coalesced row gather (float4), ~48MB traffic ~2us.

#define E_FINE   16384
#define E_COARSE 8192
#define CDIM     512
#define NTHREADS 1024
#define PER      (E_FINE / NTHREADS)   // 16 fine slots per thread
#define KPER     (E_COARSE / NTHREADS) // 8 coarse entries per thread
#define BIGD     0x40000000

#if defined(__HIP_DEVICE_COMPILE__) && __has_builtin(__builtin_amdgcn_global_load_async_to_lds_b128)
#define HAVE_ASYNC_LDS 1
#else
#define HAVE_ASYNC_LDS 0
#endif

#if HAVE_ASYNC_LDS
typedef int v4i __attribute__((vector_size(4 * sizeof(int))));
typedef v4i __attribute__((address_space(1)))* v4i_gptr;
typedef v4i __attribute__((address_space(3)))* v4i_lptr;
#endif

__global__ __launch_bounds__(NTHREADS) void nn_map_kernel(const int* __restrict__ keep_idx,
                                                          int* __restrict__ src_row) {
    __shared__ int inv[E_FINE];   // fine slot -> coarse position j, or -1
    __shared__ int buf[E_FINE];   // staging + scan buffer
    const int tid = threadIdx.x;

    // ---- stage keep_idx[0..E_COARSE) into buf[0..E_COARSE) ----
#if HAVE_ASYNC_LDS
    {
        v4i_gptr g = (v4i_gptr)(keep_idx + tid * KPER);
        v4i_lptr l = (v4i_lptr)(&buf[tid * KPER]);
        // INST_OFFSET applies to both global and LDS addresses (CDNA5 ISA)
        __builtin_amdgcn_global_load_async_to_lds_b128(g, l, 0, 0);
        __builtin_amdgcn_global_load_async_to_lds_b128(g, l, 16, 0);
    }
#else
    for (int k = 0; k < KPER; ++k) {
        int j = tid * KPER + k;
        buf[j] = keep_idx[j];
    }
#endif

    // init inverse map while async transfers are in flight
    for (int k = 0; k < PER; ++k) inv[tid + k * NTHREADS] = -1;

#if HAVE_ASYNC_LDS
#if __has_builtin(__builtin_amdgcn_s_wait_asynccnt)
    __builtin_amdgcn_s_wait_asynccnt(0);
#else
    asm volatile("s_wait_asynccnt 0" ::: "memory");
#endif
#endif
    __syncthreads();

    // ---- scatter: inv[keep_idx[j]] = j  (keep_idx unique -> no races) ----
    for (int k = 0; k < KPER; ++k) {
        int j = tid * KPER + k;
        inv[buf[j]] = j;
    }
    __syncthreads();

    // ---- prev-kept: inclusive max-scan of (kept ? i : -1) ----
    for (int k = 0; k < PER; ++k) {
        int i = tid + k * NTHREADS;
        buf[i] = (inv[i] >= 0) ? i : -1;
    }
    __syncthreads();
    for (int step = 1; step < E_FINE; step <<= 1) {
        int tmp[PER];
        for (int k = 0; k < PER; ++k) {
            int i = tid + k * NTHREADS;
            int v = buf[i];
            if (i >= step) {
                int w = buf[i - step];
                v = (w > v) ? w : v;
            }
            tmp[k] = v;
        }
        __syncthreads();
        for (int k = 0; k < PER; ++k) buf[tid + k * NTHREADS] = tmp[k];
        __syncthreads();
    }
    int prevReg[PER];
    for (int k = 0; k < PER; ++k) prevReg[k] = buf[tid + k * NTHREADS];
    __syncthreads();

    // ---- next-kept: inclusive min-scan (from the right) of (kept ? i : BIG) ----
    for (int k = 0; k < PER; ++k) {
        int i = tid + k * NTHREADS;
        buf[i] = (inv[i] >= 0) ? i : BIGD;
    }
    __syncthreads();
    for (int step = 1; step < E_FINE; step <<= 1) {
        int tmp[PER];
        for (int k = 0; k < PER; ++k) {
            int i = tid + k * NTHREADS;
            int v = buf[i];
            if (i + step < E_FINE) {
                int w = buf[i + step];
                v = (w < v) ? w : v;
            }
            tmp[k] = v;
        }
        __syncthreads();
        for (int k = 0; k < PER; ++k) buf[tid + k * NTHREADS] = tmp[k];
        __syncthreads();
    }

    // ---- combine with argmin tie-break (smaller coarse position j wins) ----
    for (int k = 0; k < PER; ++k) {
        int i = tid + k * NTHREADS;
        int ji = inv[i];
        int s;
        if (ji >= 0) {
            s = ji;                       // kept slot: its own coarse row
        } else {
            int p  = prevReg[k];          // nearest kept strictly left (-1 if none)
            int n  = buf[i];              // nearest kept strictly right (BIGD if none)
            int dl = (p >= 0)     ? (i - p) : BIGD;
            int dr = (n < E_FINE) ? (n - i) : BIGD;
            if (dl < dr)      s = inv[p];
            else if (dr < dl) s = inv[n];
            else {                        // tie: first minimal j (jnp.argmin semantics)
                int jl = inv[p], jr = inv[n];
                s = (jl < jr) ? jl : jr;
            }
        }
        src_row[i] = s;
    }
}

__global__ __launch_bounds__(256) void gather_kernel(const float* __restrict__ x_coarse,
                                                     const int* __restrict__ src_row,
                                                     float* __restrict__ out) {
    const int row = (blockIdx.x << 1) | (threadIdx.x >> 7); // 2 rows per block
    const int c4  = threadIdx.x & 127;                      // float4 lane within row
    const int s   = src_row[row];
    const float4* in = (const float4*)x_coarse + (size_t)s * (CDIM / 4);
    float4*       op = (float4*)out          + (size_t)row * (CDIM / 4);
    __builtin_prefetch(in + c4, 0, 0);   // global_prefetch_b8
    op[c4] = in[c4];
}

extern "C" void kernel_launch(void* const* d_in, const int* in_sizes, int n_in,
                              void* d_out, int out_size, void* d_ws, size_t ws_size,
                              hipStream_t stream) {
    const float* x_coarse = (const float*)d_in[0];
    const int*   keep_idx = (const int*)d_in[1];
    int*         src_row  = (int*)d_ws;            // E_FINE ints = 64 KB scratch
    float*       out      = (float*)d_out;

    nn_map_kernel<<<1, NTHREADS, 0, stream>>>(keep_idx, src_row);
    gather_kernel<<<E_FINE / 2, 256, 0, stream>>>(x_coarse, src_row, out);
}